// Net_15762529976718
// MI455X (gfx1250) — compile-verified
//
#include <hip/hip_runtime.h>
#include <math.h>

typedef __attribute__((ext_vector_type(2))) float v2f;
typedef __attribute__((ext_vector_type(8))) float v8f;

#define F_IN 512
#define HID  16
#define NCLS 40

// deg := 1.0 (self loop), agg1 := 0
__global__ void k_init(float* __restrict__ deg, float* __restrict__ agg1, int n) {
  int t = blockIdx.x * blockDim.x + threadIdx.x;
  if (t < n * HID) {
    agg1[t] = 0.f;
    if ((t & (HID - 1)) == 0) deg[t >> 4] = 1.0f;
  }
}

__global__ void k_deg(const int* __restrict__ dst, float* __restrict__ deg, int n_edges) {
  int e = blockIdx.x * blockDim.x + threadIdx.x;
  if (e < n_edges) atomicAdd(&deg[dst[e]], 1.0f);
}

__global__ void k_dinv(float* __restrict__ deg, int n) {
  int i = blockIdx.x * blockDim.x + threadIdx.x;
  if (i < n) deg[i] = rsqrtf(deg[i]);   // deg >= 1 always (self loops)
}

// h[N,16] = x[N,512] @ W1[512,16] via V_WMMA_F32_16X16X4_F32.
// One wave = one 16-row tile; K marched 4 at a time (128 chained WMMAs).
__global__ void k_gemm1(const float* __restrict__ x, const float* __restrict__ W1,
                        float* __restrict__ h, int n_tiles, int n_rows) {
  int wave = (int)((blockIdx.x * blockDim.x + threadIdx.x) >> 5);
  int lane = threadIdx.x & 31;
  if (wave >= n_tiles) return;          // uniform per wave -> EXEC stays all-ones
  int row = lane & 15;                  // M (for A) / N (for B,D)
  int hi  = lane >> 4;
  int grow = wave * 16 + row;
  if (grow >= n_rows) grow = n_rows - 1;        // pad tail tile by clamping
  const float* __restrict__ xrow = x + (size_t)grow * F_IN;

  v8f c = {};
  #pragma unroll 8
  for (int k = 0; k < F_IN; k += 4) {
    v2f a, b;
    // A 16x4: lane(row,hi), VGPR v holds A[row][2*hi + v]; x is stream-once -> NT
    a.x = __builtin_nontemporal_load(&xrow[k + 2 * hi + 0]);
    a.y = __builtin_nontemporal_load(&xrow[k + 2 * hi + 1]);
    // B 4x16: VGPR v holds B[2*hi + v][col=row]; W1 is hot -> regular loads
    b.x = W1[(k + 2 * hi + 0) * HID + row];
    b.y = W1[(k + 2 * hi + 1) * HID + row];
    c = __builtin_amdgcn_wmma_f32_16x16x4_f32(false, a, false, b, (short)0, c,
                                              false, false);
  }
  // D 16x16: VGPR v holds D[v + 8*hi][row]
  if (wave * 16 + 16 <= n_rows) {       // full tile: uniform branch, no exec churn
    float* __restrict__ hb = h + (size_t)(wave * 16 + 8 * hi) * HID + row;
    #pragma unroll
    for (int v = 0; v < 8; ++v) hb[v * HID] = c[v];
  } else {                              // tail tile only
    #pragma unroll
    for (int v = 0; v < 8; ++v) {
      int m = wave * 16 + v + 8 * hi;
      if (m < n_rows) h[(size_t)m * HID + row] = c[v];
    }
  }
}

// 16 lanes per edge: agg[dst][j] += h[src][j] * dinv[src]*dinv[dst]
__global__ void k_edge(const int* __restrict__ src, const int* __restrict__ dst,
                       const float* __restrict__ dinv, const float* __restrict__ hsrc,
                       float* __restrict__ agg, int n_edges) {
  int t = blockIdx.x * blockDim.x + threadIdx.x;
  int e = t >> 4, j = t & 15;
  if (e >= n_edges) return;
  int s = src[e], d = dst[e];
  float nrm = dinv[s] * dinv[d];
  atomicAdd(&agg[d * HID + j], hsrc[s * HID + j] * nrm);
}

// hrelu (in-place over agg1) = relu(agg1 + h*dinv^2 + b1); agg2 := 0
__global__ void k_post1(float* __restrict__ agg1, const float* __restrict__ h,
                        const float* __restrict__ dinv, const float* __restrict__ b1,
                        float* __restrict__ agg2, int n) {
  int t = blockIdx.x * blockDim.x + threadIdx.x;
  if (t >= n * HID) return;
  int i = t >> 4, j = t & 15;
  float di = dinv[i];
  float v = agg1[t] + h[t] * di * di + b1[j];
  agg1[t] = v > 0.f ? v : 0.f;
  agg2[t] = 0.f;
}

// out = log_softmax((agg2 + hrelu*dinv^2) @ W2 + b2)   (W2 hoisted past aggregation)
__global__ void k_post2(const float* __restrict__ agg2, const float* __restrict__ hrelu,
                        const float* __restrict__ dinv, const float* __restrict__ W2,
                        const float* __restrict__ b2, float* __restrict__ out, int n) {
  __shared__ float sW[HID * NCLS];
  __shared__ float sb[NCLS];
  for (int t = threadIdx.x; t < HID * NCLS; t += blockDim.x) sW[t] = W2[t];
  for (int t = threadIdx.x; t < NCLS; t += blockDim.x) sb[t] = b2[t];
  __syncthreads();
  int i = blockIdx.x * blockDim.x + threadIdx.x;
  if (i >= n) return;
  float di = dinv[i];
  float w = di * di;
  float r[HID];
  #pragma unroll
  for (int j = 0; j < HID; ++j) {
    int idx = i * HID + j;
    r[j] = agg2[idx] + hrelu[idx] * w;
  }
  float z[NCLS];
  float mx = -INFINITY;
  #pragma unroll
  for (int c = 0; c < NCLS; ++c) {
    float acc = sb[c];
    #pragma unroll
    for (int j = 0; j < HID; ++j) acc += r[j] * sW[j * NCLS + c];
    z[c] = acc;
    mx = fmaxf(mx, acc);
  }
  float s = 0.f;
  #pragma unroll
  for (int c = 0; c < NCLS; ++c) s += expf(z[c] - mx);
  float lse = logf(s) + mx;
  #pragma unroll
  for (int c = 0; c < NCLS; ++c)
    __builtin_nontemporal_store(z[c] - lse, &out[(size_t)i * NCLS + c]);
}

extern "C" void kernel_launch(void* const* d_in, const int* in_sizes, int n_in,
                              void* d_out, int out_size, void* d_ws, size_t ws_size,
                              hipStream_t stream) {
  const float* x  = (const float*)d_in[0];
  const int*   ei = (const int*)d_in[1];
  const float* W1 = (const float*)d_in[2];
  const float* b1 = (const float*)d_in[3];
  const float* W2 = (const float*)d_in[4];
  const float* b2 = (const float*)d_in[5];

  int n_nodes = in_sizes[0] / F_IN;
  int n_edges = in_sizes[1] / 2;
  const int* src = ei;
  const int* dst = ei + n_edges;

  char* ws = (char*)d_ws;
  size_t off = 0;
  auto take = [&](size_t nf) -> float* {
    float* p = (float*)(ws + off);
    off += (nf * sizeof(float) + 255) & ~(size_t)255;
    return p;
  };
  float* dinv = take((size_t)n_nodes);            // deg -> dinv in place
  float* h    = take((size_t)n_nodes * HID);
  float* agg1 = take((size_t)n_nodes * HID);      // later reused as hrelu
  float* agg2 = take((size_t)n_nodes * HID);

  const int T = 256;
  int nh = n_nodes * HID;
  long long et = (long long)n_edges * HID;

  k_init<<<(nh + T - 1) / T, T, 0, stream>>>(dinv, agg1, n_nodes);
  k_deg<<<(n_edges + T - 1) / T, T, 0, stream>>>(dst, dinv, n_edges);
  k_dinv<<<(n_nodes + T - 1) / T, T, 0, stream>>>(dinv, n_nodes);

  int n_tiles = (n_nodes + 15) / 16;
  k_gemm1<<<(n_tiles + 3) / 4, 128, 0, stream>>>(x, W1, h, n_tiles, n_nodes);

  k_edge<<<(unsigned)((et + T - 1) / T), T, 0, stream>>>(src, dst, dinv, h, agg1, n_edges);
  k_post1<<<(nh + T - 1) / T, T, 0, stream>>>(agg1, h, dinv, b1, agg2, n_nodes);
  k_edge<<<(unsigned)((et + T - 1) / T), T, 0, stream>>>(src, dst, dinv, agg1, agg2, n_edges);
  k_post2<<<(n_nodes + 127) / 128, 128, 0, stream>>>(agg2, agg1, dinv, W2, b2,
                                                     (float*)d_out, n_nodes);
}